// HeteroSAGE_pyg_17119739641952
// MI455X (gfx1250) — compile-verified
//
#include <hip/hip_runtime.h>
#include <hip/hip_bf16.h>

#define NP 50000
#define FDIM 128
#define HDIM 256
#define E_PPI 500000
#define NPAIR 4096

typedef float v2f __attribute__((ext_vector_type(2)));
typedef float v8f __attribute__((ext_vector_type(8)));

// ---------------------------------------------------------------- zero init
__global__ void zero_f4(float* __restrict__ p, long n4) {
    long i = (long)blockIdx.x * blockDim.x + threadIdx.x;
    long stride = (long)gridDim.x * blockDim.x;
    float4 z = make_float4(0.f, 0.f, 0.f, 0.f);
    for (; i < n4; i += stride) ((float4*)p)[i] = z;
}

// ------------------------------------------------- layer-1 scatter (F=128)
// one wave per edge: gather x[src] row (float4 per lane), atomic-add into agg[dst]
__global__ void scatter_ppi_f(const float* __restrict__ x,
                              const int* __restrict__ src,
                              const int* __restrict__ dst,
                              float* __restrict__ agg,
                              float* __restrict__ cnt) {
    int wave = (int)((blockIdx.x * blockDim.x + threadIdx.x) >> 5);
    int lane = threadIdx.x & 31;
    if (wave >= E_PPI) return;
    int s = src[wave], d = dst[wave];
    float4 v = ((const float4*)(x + (long)s * FDIM))[lane];
    float* a = agg + (long)d * FDIM + lane * 4;
    atomicAdd(a + 0, v.x);
    atomicAdd(a + 1, v.y);
    atomicAdd(a + 2, v.z);
    atomicAdd(a + 3, v.w);
    if (lane == 0) atomicAdd(cnt + d, 1.0f);
}

// ------------------------------------------------- layer-2 scatter (H=256)
__global__ void scatter_ppi_h(const float* __restrict__ x,
                              const int* __restrict__ src,
                              const int* __restrict__ dst,
                              float* __restrict__ agg) {
    int wave = (int)((blockIdx.x * blockDim.x + threadIdx.x) >> 5);
    int lane = threadIdx.x & 31;
    if (wave >= E_PPI) return;
    int s = src[wave], d = dst[wave];
    const float4* xr = (const float4*)(x + (long)s * HDIM);
    float4 v0 = xr[lane];
    float4 v1 = xr[lane + 32];
    float* a = agg + (long)d * HDIM + lane * 4;
    atomicAdd(a + 0, v0.x);
    atomicAdd(a + 1, v0.y);
    atomicAdd(a + 2, v0.z);
    atomicAdd(a + 3, v0.w);
    float* a2 = a + 128;
    atomicAdd(a2 + 0, v1.x);
    atomicAdd(a2 + 1, v1.y);
    atomicAdd(a2 + 2, v1.z);
    atomicAdd(a2 + 3, v1.w);
}

// ---------------------------------------------------------------- SAGE GEMM
// out[M,N] = relu?( (agg/max(cnt,1)) @ Wl^T + bias + xdst @ Wr^T )
// One wave per 16x16 output tile, V_WMMA_F32_16X16X4_F32, K-loop step 4.
// A layout: lane holds row (lane&15), K pair at k + ((lane>>4)*2)  -> v2f
// B layout: lane holds col (lane&15), same K pair                  -> v2f
// C/D: vgpr r, lane l -> row (r + (l>=16)*8), col (l&15)
__global__ void sage_gemm(const float* __restrict__ agg,
                          const float* __restrict__ cnt,
                          const float* __restrict__ xdst,
                          const float* __restrict__ Wl,   // [N,K] row-major
                          const float* __restrict__ bias, // [N]
                          const float* __restrict__ Wr,   // [N,K]
                          float* __restrict__ out,        // [M,N]
                          int N, int K, int do_relu) {
    int tile = blockIdx.x * (blockDim.x >> 5) + (threadIdx.x >> 5);
    int lane = threadIdx.x & 31;
    int nct = N >> 4;
    int rt = tile / nct;
    int ct = tile - rt * nct;
    int m = (rt << 4) + (lane & 15);
    int n = (ct << 4) + (lane & 15);
    int kh = (lane >> 4) << 1;  // 0 or 2

    float inv = 1.0f / fmaxf(cnt[m], 1.0f);
    const float* arow  = agg  + (long)m * K + kh;
    const float* xrow  = xdst + (long)m * K + kh;
    const float* wlrow = Wl   + (long)n * K + kh;
    const float* wrrow = Wr   + (long)n * K + kh;

    v8f c = {};
    for (int k = 0; k < K; k += 4) {
        v2f a = *(const v2f*)(arow + k);
        a *= inv;  // fold mean scale into A
        v2f b = *(const v2f*)(wlrow + k);
        c = __builtin_amdgcn_wmma_f32_16x16x4_f32(false, a, false, b,
                                                  (short)0, c, false, false);
    }
    for (int k = 0; k < K; k += 4) {
        v2f a = *(const v2f*)(xrow + k);
        v2f b = *(const v2f*)(wrrow + k);
        c = __builtin_amdgcn_wmma_f32_16x16x4_f32(false, a, false, b,
                                                  (short)0, c, false, false);
    }

    float bn = bias[n];
    int rbase = (rt << 4) + ((lane >> 4) << 3);
#pragma unroll
    for (int r = 0; r < 8; ++r) {
        float v = c[r] + bn;
        if (do_relu) v = fmaxf(v, 0.0f);
        out[(long)(rbase + r) * N + n] = v;
    }
}

// ---------------------------------------------------------------- pair head
// one wave per pair: dot(hp2[m0], W[0:128]) + dot(hp2[m1], W[128:256]) -> sigmoid
__global__ void pair_head(const float* __restrict__ hp2,
                          const float* __restrict__ linW,
                          const float* __restrict__ linb,
                          const int* __restrict__ mask,
                          float* __restrict__ out) {
    int pair = blockIdx.x * (blockDim.x >> 5) + (threadIdx.x >> 5);
    int lane = threadIdx.x & 31;
    int i1 = mask[2 * pair];
    int i2 = mask[2 * pair + 1];
    float4 a  = ((const float4*)(hp2 + (long)i1 * FDIM))[lane];
    float4 w  = ((const float4*)(linW))[lane];
    float s = a.x * w.x + a.y * w.y + a.z * w.z + a.w * w.w;
    float4 b  = ((const float4*)(hp2 + (long)i2 * FDIM))[lane];
    float4 w2 = ((const float4*)(linW + FDIM))[lane];
    s += b.x * w2.x + b.y * w2.y + b.z * w2.z + b.w * w2.w;
    for (int off = 16; off; off >>= 1) s += __shfl_down(s, off, 32);
    if (lane == 0) out[pair] = 1.0f / (1.0f + expf(-(s + linb[0])));
}

// ---------------------------------------------------------------- launcher
extern "C" void kernel_launch(void* const* d_in, const int* in_sizes, int n_in,
                              void* d_out, int out_size, void* d_ws, size_t ws_size,
                              hipStream_t stream) {
    const float* x_protein = (const float*)d_in[0];
    // a-layer ppi params (din=128, dout=256)
    const float* aWl = (const float*)d_in[11];
    const float* ab  = (const float*)d_in[12];
    const float* aWr = (const float*)d_in[13];
    // b-layer ppi params (din=256, dout=128)
    const float* bWl = (const float*)d_in[23];
    const float* bb  = (const float*)d_in[24];
    const float* bWr = (const float*)d_in[25];
    const float* linW = (const float*)d_in[26];
    const float* linb = (const float*)d_in[27];
    const int* ppi_src = (const int*)d_in[34];
    const int* ppi_dst = (const int*)d_in[35];
    const int* mask    = (const int*)d_in[36];
    float* out = (float*)d_out;

    // workspace layout (floats): [cnt NP | agg1 NP*F | agg2 NP*H | hp NP*H]
    // hp2 reuses agg1 (dead after gemm1). Total = NP*(1+F+2H)*4 = 128.2 MB
    float* ws   = (float*)d_ws;
    float* cnt  = ws;
    float* agg1 = ws + NP;
    float* agg2 = agg1 + (long)NP * FDIM;
    float* hp   = agg2 + (long)NP * HDIM;
    float* hp2  = agg1;

    // zero cnt+agg1+agg2 (contiguous prefix): NP*(1+128+256) floats
    long nz = (long)NP * (1 + FDIM + HDIM);
    zero_f4<<<2048, 256, 0, stream>>>(ws, nz / 4);

    // layer 1: scatter-mean over ppi, then hp = relu(mean@aWl^T + ab + x@aWr^T)
    scatter_ppi_f<<<(E_PPI * 32) / 256, 256, 0, stream>>>(x_protein, ppi_src, ppi_dst,
                                                          agg1, cnt);
    {
        int tiles = (NP / 16) * (HDIM / 16);  // 50000
        sage_gemm<<<tiles / 8, 256, 0, stream>>>(agg1, cnt, x_protein, aWl, ab, aWr,
                                                 hp, HDIM, FDIM, 1);
    }

    // layer 2: scatter-mean of hp over same edges (counts identical), then hp2
    scatter_ppi_h<<<(E_PPI * 32) / 256, 256, 0, stream>>>(hp, ppi_src, ppi_dst, agg2);
    {
        int tiles = (NP / 16) * (FDIM / 16);  // 25000
        sage_gemm<<<tiles / 8, 256, 0, stream>>>(agg2, cnt, hp, bWl, bb, bWr,
                                                 hp2, FDIM, HDIM, 0);
    }

    // pair readout head
    pair_head<<<NPAIR / 8, 256, 0, stream>>>(hp2, linW, linb, mask, out);
}